// EmergentNeuralNetwork_3212635538184
// MI455X (gfx1250) — compile-verified
//
#include <hip/hip_runtime.h>
#include <math.h>

// ---------------------------------------------------------------------------
// EmergentNeuralNetwork on MI455X (gfx1250, wave32).
//
// out[b,o] = tanh( h0*w_ho[0,o] + h1*w_ho[1,o] + h2*w_ho[2,o] + Cst[o] - 0.5 )
//   h_c    = tanh( (x @ w_in)[b,c] - thr_h[c] ),  c = 0..2
//   Cst[o] = sum_{j=3..63} tanh(-thr_h[j]) * w_ho[j,o]
//
// Bandwidth-bound: 128MB of x @ 23.3 TB/s => ~5.5us floor. K split 8 ways
// (4096 waves) for memory-level parallelism; f32 WMMA 16x16x4 keeps exact
// reference precision. B operand comes from a zero-padded [k][4] LDS image of
// w_in so every lane's fetch is unconditional (no exec-mask churn), and each
// K-chunk batches 16 independent global b64 loads before the 16 WMMAs.
// Deterministic: each partial-sum slot written exactly once, fixed-order sums.
// ---------------------------------------------------------------------------

typedef float v2f __attribute__((ext_vector_type(2)));
typedef float v4f __attribute__((ext_vector_type(4)));
typedef float v8f __attribute__((ext_vector_type(8)));

#define BATCH   8192
#define IN      4096
#define NH      64
#define NC      3          // inputs feed only first 3 hidden neurons
#define NO      8
#define KSPLIT  8
#define KLEN    (IN / KSPLIT)   // 512 per wave
#define UNROLL  16              // K elements per chunk = 4*UNROLL = 64

// Kernel 1: partial GEMM  part[kidx][row][c] = sum over K-slice of x*w_in.
// Block = 256 threads = 8 waves. Block b owns rows [16b, 16b+16);
// wave w of the block owns K-slice [512w, 512w+512).
__global__ __launch_bounds__(256)
void enn_gemm_partial(const float* __restrict__ x,
                      const float* __restrict__ w_in,
                      float* __restrict__ part) {
    // w_in staged as [k][4] with column 3 == 0 (64KB of 320KB WGP LDS).
    // Lanes with B-column >= 3 read the zero column -> branch-free B fetch.
    __shared__ __align__(16) float lds_w[IN * 4];

    const int tid = threadIdx.x;
    for (int k = tid; k < IN; k += 256) {
        v4f v;
        v.x = w_in[k * NC + 0];
        v.y = w_in[k * NC + 1];
        v.z = w_in[k * NC + 2];
        v.w = 0.0f;
        *reinterpret_cast<v4f*>(&lds_w[k * 4]) = v;
    }
    __syncthreads();

    const int wave = tid >> 5;              // = kidx (K-slice id)
    const int lane = tid & 31;
    const int hl   = lane >> 4;             // half-wave: K pair {0,1} vs {2,3}
    const int rl   = lane & 15;             // row-within-tile / B column
    const int rl4  = (rl < NC) ? rl : NC;   // padded B column (3 -> zeros)

    const float* xrow  = x + (size_t)(blockIdx.x * 16 + rl) * IN;
    const int    kbase = wave * KLEN;

    v8f acc = {};   // f32 16x16 accumulator tile (8 VGPRs)

    for (int k0 = kbase; k0 < kbase + KLEN; k0 += 4 * UNROLL) {
        v2f a[UNROLL], b[UNROLL];
        // Batch A loads: 16 independent global_load_b64 in flight per lane.
        #pragma unroll
        for (int u = 0; u < UNROLL; ++u) {
            const int kk = k0 + 4 * u + 2 * hl;
            a[u] = *reinterpret_cast<const v2f*>(xrow + kk);
        }
        // Batch B fetches from LDS: unconditional, all lanes active.
        #pragma unroll
        for (int u = 0; u < UNROLL; ++u) {
            const int kk = k0 + 4 * u + 2 * hl;
            b[u].x = lds_w[(kk    ) * 4 + rl4];
            b[u].y = lds_w[(kk + 1) * 4 + rl4];
        }
        // 16 chained f32 WMMAs (D = A(16x4) x B(4x16) + C).
        #pragma unroll
        for (int u = 0; u < UNROLL; ++u) {
            acc = __builtin_amdgcn_wmma_f32_16x16x4_f32(
                      false, a[u], false, b[u], (short)0, acc, false, false);
        }
    }

    // C/D layout: VGPR r = (M=r, N=lane) for lanes 0-15, (M=r+8, N=lane-16)
    // for lanes 16-31. Lanes with N<3 scatter their column of 8 rows.
    if (rl < NC) {
        float* dstp = part +
            ((size_t)wave * BATCH + blockIdx.x * 16 + 8 * hl) * NC + rl;
        #pragma unroll
        for (int r = 0; r < 8; ++r)
            dstp[(size_t)r * NC] = acc[r];
    }
}

// Kernel 2: reduce the 8 K-partials, apply tanh epilogue, add the constant
// contribution of the 61 unconnected hidden neurons. One thread per row.
__global__ __launch_bounds__(256)
void enn_epilogue(const float* __restrict__ part,
                  const float* __restrict__ w_ho,
                  const float* __restrict__ thr_h,
                  float* __restrict__ out) {
    __shared__ float cst[NO];
    if (threadIdx.x < NO) {
        float s = 0.0f;
        for (int j = NC; j < NH; ++j)
            s += tanhf(-thr_h[j]) * w_ho[j * NO + threadIdx.x];
        cst[threadIdx.x] = s - 0.5f;      // fold output threshold
    }
    __syncthreads();

    const int row = blockIdx.x * blockDim.x + threadIdx.x;   // 8192 = 32*256
    float h[NC];
    #pragma unroll
    for (int c = 0; c < NC; ++c) {
        float s = 0.0f;
        #pragma unroll
        for (int kidx = 0; kidx < KSPLIT; ++kidx)
            s += part[((size_t)kidx * BATCH + row) * NC + c];
        h[c] = tanhf(s - thr_h[c]);
    }
    #pragma unroll
    for (int o = 0; o < NO; ++o) {
        float s = cst[o] + h[0] * w_ho[o]
                         + h[1] * w_ho[NO + o]
                         + h[2] * w_ho[2 * NO + o];
        out[(size_t)row * NO + o] = tanhf(s);
    }
}

extern "C" void kernel_launch(void* const* d_in, const int* in_sizes, int n_in,
                              void* d_out, int out_size, void* d_ws, size_t ws_size,
                              hipStream_t stream) {
    (void)in_sizes; (void)n_in; (void)out_size; (void)ws_size;
    const float* x     = (const float*)d_in[0];   // [8192,4096]
    const float* w_in  = (const float*)d_in[1];   // [4096,3]
    const float* w_ho  = (const float*)d_in[2];   // [64,8]
    const float* thr_h = (const float*)d_in[3];   // [64]
    float* out  = (float*)d_out;                  // [8192,8]
    float* part = (float*)d_ws;                   // KSPLIT*BATCH*NC f32 = 768KB

    enn_gemm_partial<<<BATCH / 16, 256, 0, stream>>>(x, w_in, part);
    enn_epilogue<<<BATCH / 256, 256, 0, stream>>>(part, w_ho, thr_h, out);
}